// TransformerNetModel_Diffulex_2757369004493
// MI455X (gfx1250) — compile-verified
//
#include <hip/hip_runtime.h>
#include <math.h>
#include <stdint.h>

// ---------------------------------------------------------------------------
// Transformer decoder layer for MI455X (gfx1250), fp32, WMMA 16x16x4_f32 path,
// async-to-LDS staging (ASYNCcnt). T=2048, B=2, C=128, H=8, Dh=16, FFN=1024.
// ---------------------------------------------------------------------------

typedef float v2f __attribute__((ext_vector_type(2)));
typedef float v8f __attribute__((ext_vector_type(8)));

#define TT 2048
#define BB 2
#define CC 128
#define HH 8
#define DH 16
#define FF 1024
#define NN (TT * BB)       // 4096 token rows
#define ROWSTR (BB * CC)   // 256 floats between consecutive t for fixed b

__device__ __forceinline__ v8f wmma_f32(v2f a, v2f b, v8f c) {
  return __builtin_amdgcn_wmma_f32_16x16x4_f32(
      /*neg_a=*/false, a, /*neg_b=*/false, b,
      /*c_mod=*/(short)0, c, /*reuse_a=*/false, /*reuse_b=*/false);
}

// LDS byte offset from a generic pointer to a __shared__ object: the LDS
// aperture keeps the offset in the low 32 bits of the flat address.
__device__ __forceinline__ uint32_t lds_off(const void* p) {
  return (uint32_t)(uintptr_t)p;
}

// CDNA5 async copy: 16 bytes global -> LDS, tracked with ASYNCcnt.
__device__ __forceinline__ void async_b128(uint32_t lds_addr, const void* gaddr) {
  asm volatile("global_load_async_to_lds_b128 %0, %1, off"
               :: "v"(lds_addr), "v"(gaddr)
               : "memory");
}

__device__ __forceinline__ void async_wait0() {
  asm volatile("s_wait_asynccnt 0" ::: "memory");
}

// ---------------------------------------------------------------------------
// GEMM: out[N x M] = A[N x K] @ W[K x M] (+bias, fused epilogue).
// Block: 256 thr = 8 waves as (4 row-strips x 2 col-tiles); block tile is
// 64 rows x 128 cols; each wave owns a 16x64 tile (4 WMMA accumulators).
// K is tiled in chunks of 128; A chunk (64x128) and W chunk (128x128) staged
// in LDS via async-to-LDS b128.  grid = (N/64, M/128).
// op: 0 -> (acc+bias)*scale ; 1 -> gelu_exact(acc+bias) ; 2 -> acc+bias+resid
// ---------------------------------------------------------------------------
__global__ __launch_bounds__(256) void gemm_wmma(
    const float* __restrict__ A, const float* __restrict__ W,
    const float* __restrict__ bias, const float* __restrict__ resid,
    float* __restrict__ out, int K, int M, float scale, int op) {
  __shared__ float As[64 * 128];   // 32 KB
  __shared__ float Ws[128 * 128];  // 64 KB

  const int rowbase = blockIdx.x * 64;
  const int colgrp  = blockIdx.y * 128;
  const int tid  = threadIdx.x;
  const int wave = tid >> 5;
  const int lane = tid & 31;
  const int mn = lane & 15;
  const int hl = lane >> 4;
  const int rs = wave >> 1;          // row strip 0..3
  const int ct = wave & 1;           // col tile  0..1 (64 cols each)
  const int colbase = colgrp + ct * 64;

  v8f acc[4] = {};

  for (int kc = 0; kc < K; kc += 128) {
    // Stage A chunk: 64 x 128 floats = 2048 x b128.
    for (int i = tid; i < 2048; i += 256) {
      const int r = i >> 5, c = (i & 31) * 4;
      async_b128(lds_off(&As[r * 128 + c]), A + (rowbase + r) * K + kc + c);
    }
    // Stage W chunk: 128 x 128 floats = 4096 x b128.
    for (int i = tid; i < 4096; i += 256) {
      const int r = i >> 5, c = (i & 31) * 4;
      async_b128(lds_off(&Ws[r * 128 + c]), W + (kc + r) * M + colgrp + c);
    }
    async_wait0();
    __syncthreads();

    for (int k0 = 0; k0 < 128; k0 += 4) {
      const int ka = k0 + 2 * hl;  // reg r holds k = ka + r
      v2f a;
      a.x = As[(rs * 16 + mn) * 128 + ka];
      a.y = As[(rs * 16 + mn) * 128 + ka + 1];
#pragma unroll
      for (int j = 0; j < 4; ++j) {
        v2f b;
        b.x = Ws[ka * 128 + ct * 64 + j * 16 + mn];
        b.y = Ws[(ka + 1) * 128 + ct * 64 + j * 16 + mn];
        acc[j] = wmma_f32(a, b, acc[j]);
      }
    }
    __syncthreads();
  }

#pragma unroll
  for (int j = 0; j < 4; ++j) {
    const int col = colbase + j * 16 + mn;
    const float bv = bias[col];
#pragma unroll
    for (int r = 0; r < 8; ++r) {
      const int row = rowbase + rs * 16 + r + 8 * hl;
      float v = acc[j][r] + bv;
      if (op == 0) {
        v *= scale;
      } else if (op == 1) {
        v = 0.5f * v * (1.0f + erff(v * 0.70710678118654752f));  // exact GELU
      } else {  // op == 2
        v += resid[row * M + col];
      }
      out[row * M + col] = v;
    }
  }
}

// ---------------------------------------------------------------------------
// Flash attention, fp32 WMMA.  Q,K,V,O logically [T, Dh] per (b,h), row
// stride ROWSTR, base offset b*C + h*Dh.  Block = 128 queries (8 waves x 16).
// K/V staged in LDS in 128-key tiles via async-to-LDS; per-16-key subtile:
// 4 WMMAs (QK^T) -> online softmax -> P via LDS -> 4 WMMAs (P*V).
// grid = (T/128, B*H).  Q is pre-scaled by Dh^-0.5.
// ---------------------------------------------------------------------------
__global__ __launch_bounds__(256) void flash_attn(
    const float* __restrict__ Q, const float* __restrict__ K,
    const float* __restrict__ V, float* __restrict__ O, int S, int causal) {
  __shared__ float Ks[128 * 16];
  __shared__ float Vs[128 * 16];
  __shared__ float Ps[8 * 16 * 16];  // per-wave 16x16 P tile

  const int bh = blockIdx.y;
  const int b = bh / HH, h = bh - b * HH;
  const int base = b * CC + h * DH;
  const float* Qb = Q + base;
  const float* Kb = K + base;
  const float* Vb = V + base;
  float* Ob = O + base;

  const int tid = threadIdx.x;
  const int wave = tid >> 5;
  const int lane = tid & 31;
  const int mn = lane & 15;
  const int hl = lane >> 4;
  const int qbase = blockIdx.x * 128 + wave * 16;

  // Q fragment (A-matrix layout, K = Dh = 16 -> 4 chunks).
  v2f qf[4];
#pragma unroll
  for (int kk = 0; kk < 4; ++kk) {
    const int kq = 4 * kk + 2 * hl;
    qf[kk].x = Qb[(qbase + mn) * ROWSTR + kq];
    qf[kk].y = Qb[(qbase + mn) * ROWSTR + kq + 1];
  }

  float mi[8], li[8];
#pragma unroll
  for (int r = 0; r < 8; ++r) { mi[r] = -1e30f; li[r] = 0.0f; }
  v8f o = {};

  const int ntiles = causal ? (blockIdx.x + 1) : (S / 128);
  for (int t = 0; t < ntiles; ++t) {
    const int kt0 = t * 128;
    // Async-stage K and V tiles: 128 rows x 16 floats = 512 x b128 each.
    for (int i = tid; i < 512; i += 256) {
      const int rr = i >> 2, c4 = (i & 3) * 4;
      async_b128(lds_off(&Ks[rr * 16 + c4]), Kb + (kt0 + rr) * ROWSTR + c4);
      async_b128(lds_off(&Vs[rr * 16 + c4]), Vb + (kt0 + rr) * ROWSTR + c4);
    }
    async_wait0();
    __syncthreads();

    for (int sub = 0; sub < 8; ++sub) {
      const int kb0 = kt0 + sub * 16;
      if (causal && kb0 > qbase + 15) break;  // wave-uniform early exit

      // S = Q * K^T  (16 queries x 16 keys)
      v8f s = {};
#pragma unroll
      for (int kk = 0; kk < 4; ++kk) {
        const int kd = 4 * kk + 2 * hl;
        v2f bf;
        bf.x = Ks[(sub * 16 + mn) * 16 + kd];
        bf.y = Ks[(sub * 16 + mn) * 16 + kd + 1];
        s = wmma_f32(qf[kk], bf, s);
      }

      // Online softmax per row (row r + 8*hl stays inside one lane-half).
#pragma unroll
      for (int r = 0; r < 8; ++r) {
        float sv = s[r];
        if (causal) {
          const int qrow = qbase + r + 8 * hl;
          if (kb0 + mn > qrow) sv += -1e9f;  // additive mask, as reference
        }
        float tmax = sv;
        tmax = fmaxf(tmax, __shfl_xor(tmax, 1));
        tmax = fmaxf(tmax, __shfl_xor(tmax, 2));
        tmax = fmaxf(tmax, __shfl_xor(tmax, 4));
        tmax = fmaxf(tmax, __shfl_xor(tmax, 8));
        const float mnew = fmaxf(mi[r], tmax);
        const float alpha = __expf(mi[r] - mnew);
        const float p = __expf(sv - mnew);
        float rsum = p;
        rsum += __shfl_xor(rsum, 1);
        rsum += __shfl_xor(rsum, 2);
        rsum += __shfl_xor(rsum, 4);
        rsum += __shfl_xor(rsum, 8);
        li[r] = li[r] * alpha + rsum;
        mi[r] = mnew;
        o[r] *= alpha;
        Ps[wave * 256 + (r + 8 * hl) * 16 + mn] = p;  // wave-private region
      }

      // O += P * V  (A-frags of P re-read from LDS; compiler tracks DScnt).
#pragma unroll
      for (int kk = 0; kk < 4; ++kk) {
        const int kl = 4 * kk + 2 * hl;
        v2f a2, bf;
        a2.x = Ps[wave * 256 + mn * 16 + kl];
        a2.y = Ps[wave * 256 + mn * 16 + kl + 1];
        bf.x = Vs[(sub * 16 + kl) * 16 + mn];
        bf.y = Vs[(sub * 16 + kl + 1) * 16 + mn];
        o = wmma_f32(a2, bf, o);
      }
    }
    __syncthreads();
  }

#pragma unroll
  for (int r = 0; r < 8; ++r) {
    const int row = qbase + r + 8 * hl;
    Ob[row * ROWSTR + mn] = o[r] / li[r];
  }
}

// ---------------------------------------------------------------------------
// Row LayerNorm over C=128: one wave per row, float4 loads, shuffle reduce.
// ---------------------------------------------------------------------------
__global__ __launch_bounds__(256) void layernorm_rows(
    const float* __restrict__ X, const float* __restrict__ g,
    const float* __restrict__ bta, float* __restrict__ out) {
  const int tid = threadIdx.x;
  const int wave = tid >> 5, lane = tid & 31;
  const int row = blockIdx.x * 8 + wave;

  const float4 xv = reinterpret_cast<const float4*>(X + row * CC)[lane];
  float s = xv.x + xv.y + xv.z + xv.w;
  float ss = xv.x * xv.x + xv.y * xv.y + xv.z * xv.z + xv.w * xv.w;
#pragma unroll
  for (int m = 1; m < 32; m <<= 1) {
    s += __shfl_xor(s, m);
    ss += __shfl_xor(ss, m);
  }
  const float mean = s * (1.0f / CC);
  const float var = ss * (1.0f / CC) - mean * mean;
  const float rstd = rsqrtf(var + 1e-5f);

  const float4 gv = reinterpret_cast<const float4*>(g)[lane];
  const float4 bv = reinterpret_cast<const float4*>(bta)[lane];
  float4 ov;
  ov.x = (xv.x - mean) * rstd * gv.x + bv.x;
  ov.y = (xv.y - mean) * rstd * gv.y + bv.y;
  ov.z = (xv.z - mean) * rstd * gv.z + bv.z;
  ov.w = (xv.w - mean) * rstd * gv.w + bv.w;
  reinterpret_cast<float4*>(out + row * CC)[lane] = ov;
}

// ---------------------------------------------------------------------------
extern "C" void kernel_launch(void* const* d_in, const int* in_sizes, int n_in,
                              void* d_out, int out_size, void* d_ws, size_t ws_size,
                              hipStream_t stream) {
  (void)in_sizes; (void)n_in; (void)out_size; (void)ws_size;

  const float* x     = (const float*)d_in[0];
  const float* enc   = (const float*)d_in[1];
  const float* sa_wq = (const float*)d_in[2];  const float* sa_bq = (const float*)d_in[3];
  const float* sa_wk = (const float*)d_in[4];  const float* sa_bk = (const float*)d_in[5];
  const float* sa_wv = (const float*)d_in[6];  const float* sa_bv = (const float*)d_in[7];
  const float* sa_wo = (const float*)d_in[8];  const float* sa_bo = (const float*)d_in[9];
  const float* ln1_g = (const float*)d_in[10]; const float* ln1_b = (const float*)d_in[11];
  const float* ca_wq = (const float*)d_in[12]; const float* ca_bq = (const float*)d_in[13];
  const float* ca_wk = (const float*)d_in[14]; const float* ca_bk = (const float*)d_in[15];
  const float* ca_wv = (const float*)d_in[16]; const float* ca_bv = (const float*)d_in[17];
  const float* ca_wo = (const float*)d_in[18]; const float* ca_bo = (const float*)d_in[19];
  const float* ln2_g = (const float*)d_in[20]; const float* ln2_b = (const float*)d_in[21];
  const float* fc1_w = (const float*)d_in[22]; const float* fc1_b = (const float*)d_in[23];
  const float* fc2_w = (const float*)d_in[24]; const float* fc2_b = (const float*)d_in[25];
  const float* ln3_g = (const float*)d_in[26]; const float* ln3_b = (const float*)d_in[27];

  // Workspace carve-out (floats): 7 * N*C + N*FFN = 30 MB.
  float* ws = (float*)d_ws;
  float* qb = ws;
  float* kb = qb + (size_t)NN * CC;
  float* vb = kb + (size_t)NN * CC;
  float* ab = vb + (size_t)NN * CC;
  float* yb = ab + (size_t)NN * CC;
  float* x1 = yb + (size_t)NN * CC;
  float* x2 = x1 + (size_t)NN * CC;
  float* hb = x2 + (size_t)NN * CC;  // N x FFN

  const dim3 blk(256);
  const dim3 gP(NN / 64, CC / 128);   // 64 x 1 (projection / out-proj / fc2)
  const dim3 gF1(NN / 64, FF / 128);  // 64 x 8 (fc1)
  const dim3 gAt(TT / 128, BB * HH);  // 16 x 16 (attention)
  const dim3 gLN(NN / 8);             // 512 (layernorm)
  const float qscale = 0.25f;         // Dh^-0.5, Dh=16

  // ---- self-attention block ----
  gemm_wmma<<<gP, blk, 0, stream>>>(x, sa_wq, sa_bq, nullptr, qb, CC, CC, qscale, 0);
  gemm_wmma<<<gP, blk, 0, stream>>>(x, sa_wk, sa_bk, nullptr, kb, CC, CC, 1.0f, 0);
  gemm_wmma<<<gP, blk, 0, stream>>>(x, sa_wv, sa_bv, nullptr, vb, CC, CC, 1.0f, 0);
  flash_attn<<<gAt, blk, 0, stream>>>(qb, kb, vb, ab, TT, 1);
  gemm_wmma<<<gP, blk, 0, stream>>>(ab, sa_wo, sa_bo, x, yb, CC, CC, 1.0f, 2);
  layernorm_rows<<<gLN, blk, 0, stream>>>(yb, ln1_g, ln1_b, x1);

  // ---- cross-attention block ----
  gemm_wmma<<<gP, blk, 0, stream>>>(x1, ca_wq, ca_bq, nullptr, qb, CC, CC, qscale, 0);
  gemm_wmma<<<gP, blk, 0, stream>>>(enc, ca_wk, ca_bk, nullptr, kb, CC, CC, 1.0f, 0);
  gemm_wmma<<<gP, blk, 0, stream>>>(enc, ca_wv, ca_bv, nullptr, vb, CC, CC, 1.0f, 0);
  flash_attn<<<gAt, blk, 0, stream>>>(qb, kb, vb, ab, TT, 0);
  gemm_wmma<<<gP, blk, 0, stream>>>(ab, ca_wo, ca_bo, x1, yb, CC, CC, 1.0f, 2);
  layernorm_rows<<<gLN, blk, 0, stream>>>(yb, ln2_g, ln2_b, x2);

  // ---- FFN block ----
  gemm_wmma<<<gF1, blk, 0, stream>>>(x2, fc1_w, fc1_b, nullptr, hb, CC, FF, 1.0f, 1);
  gemm_wmma<<<gP, blk, 0, stream>>>(hb, fc2_w, fc2_b, x2, yb, FF, CC, 1.0f, 2);
  layernorm_rows<<<gLN, blk, 0, stream>>>(yb, ln3_g, ln3_b, (float*)d_out);
}